// ProlongationMLP_63136019251922
// MI455X (gfx1250) — compile-verified
//
#include <hip/hip_runtime.h>
#include <hip/hip_bf16.h>
#include <math.h>

typedef __attribute__((ext_vector_type(2))) float v2f;
typedef __attribute__((ext_vector_type(8))) float v8f;

#define B_SZ     32
#define NNODES   2048
#define HIDDEN   2048
#define RANK     32
#define FLAT_IN  32768   // NNODES * IN_DIM
#define FLAT_OUT 65536   // NNODES * RANK

// D = A(16x4 f32) * B(4x16 f32) + C(16x16 f32)  -> v_wmma_f32_16x16x4_f32
__device__ __forceinline__ v8f wmma4(v2f a, v2f b, v8f c) {
  return __builtin_amdgcn_wmma_f32_16x16x4_f32(false, a, false, b, (short)0, c,
                                               false, false);
}

__device__ __forceinline__ float gelu_exact(float v) {
  return 0.5f * v * (1.0f + erff(v * 0.70710678118654752440f));
}

// Non-temporal scalar load: W1/W2 are 768 MB total (> 192 MB L2) and read
// exactly once -> stream them NT so L2 keeps x/h/y resident instead.
__device__ __forceinline__ float ldg_nt(const float* p) {
  return __builtin_nontemporal_load(p);
}

// ---------------------------------------------------------------------------
// Kernel 1: h = gelu(x[32,32768] @ W1[32768,2048] + b1)
// grid = 128 (16-col tiles of HIDDEN), block = 128 (4 waves split K 4 ways)
// ---------------------------------------------------------------------------
__global__ __launch_bounds__(128) void k_gemm1_gelu(
    const float* __restrict__ x, const float* __restrict__ W1,
    const float* __restrict__ b1, float* __restrict__ h) {
  __shared__ float red[4][512];
  const int tid  = threadIdx.x;
  const int w    = tid >> 5, lane = tid & 31;
  const int nbase = blockIdx.x * 16;
  const int m    = lane & 15;          // A-frag row (lanes 0-15 / 16-31 same M)
  const int kk   = (lane >> 4) * 2;    // K sub-offset: VGPR0 holds K=kk, VGPR1 K=kk+1
  const int ncol = nbase + (lane & 15);
  const int kstart = w * (FLAT_IN / 4);
  const int kend   = kstart + (FLAT_IN / 4);

  v8f acc0 = {}; v8f acc1 = {};
  const float* xr0 = x + (size_t)m * FLAT_IN;
  const float* xr1 = x + (size_t)(m + 16) * FLAT_IN;

  for (int k = kstart; k < kend; k += 4) {
    v2f a0 = *(const v2f*)(xr0 + k + kk);          // x rows 0..15 (L2-hot)
    v2f a1 = *(const v2f*)(xr1 + k + kk);          // x rows 16..31
    v2f b;
    b.x = ldg_nt(&W1[(size_t)(k + kk) * HIDDEN + ncol]);      // streamed NT
    b.y = ldg_nt(&W1[(size_t)(k + kk + 1) * HIDDEN + ncol]);
    acc0 = wmma4(a0, b, acc0);
    acc1 = wmma4(a1, b, acc1);
  }

  #pragma unroll
  for (int r = 0; r < 8; ++r) {
    red[w][r * 32 + lane]       = acc0[r];
    red[w][256 + r * 32 + lane] = acc1[r];
  }
  __syncthreads();
  for (int idx = tid; idx < 512; idx += 128) {
    float s = red[0][idx] + red[1][idx] + red[2][idx] + red[3][idx];
    int tile = idx >> 8, r = (idx >> 5) & 7, L = idx & 31;
    int mm = tile * 16 + ((L < 16) ? r : r + 8);
    int nn = nbase + (L & 15);
    h[mm * HIDDEN + nn] = gelu_exact(s + b1[nn]);
  }
}

// ---------------------------------------------------------------------------
// Kernel 2: y = h[32,2048] @ W2[2048,65536] + b2   (y lives in d_out)
// grid = 4096 (16-col tiles of FLAT_OUT), block = 128 (4 waves split K)
// ---------------------------------------------------------------------------
__global__ __launch_bounds__(128) void k_gemm2(
    const float* __restrict__ h, const float* __restrict__ W2,
    const float* __restrict__ b2, float* __restrict__ y) {
  __shared__ float red[4][512];
  const int tid  = threadIdx.x;
  const int w    = tid >> 5, lane = tid & 31;
  const int nbase = blockIdx.x * 16;
  const int m    = lane & 15;
  const int kk   = (lane >> 4) * 2;
  const int ncol = nbase + (lane & 15);
  const int kstart = w * (HIDDEN / 4);
  const int kend   = kstart + (HIDDEN / 4);

  v8f acc0 = {}; v8f acc1 = {};
  const float* hr0 = h + (size_t)m * HIDDEN;
  const float* hr1 = h + (size_t)(m + 16) * HIDDEN;

  for (int k = kstart; k < kend; k += 4) {
    if (k + 64 < kend)  // stream-ahead prefetch of W2 rows (global_prefetch_b8)
      __builtin_prefetch(&W2[(size_t)(k + 64 + kk) * FLAT_OUT + ncol], 0, 1);
    v2f a0 = *(const v2f*)(hr0 + k + kk);          // h is 256 KB, L2-hot
    v2f a1 = *(const v2f*)(hr1 + k + kk);
    v2f b;
    b.x = ldg_nt(&W2[(size_t)(k + kk) * FLAT_OUT + ncol]);    // streamed NT
    b.y = ldg_nt(&W2[(size_t)(k + kk + 1) * FLAT_OUT + ncol]);
    acc0 = wmma4(a0, b, acc0);
    acc1 = wmma4(a1, b, acc1);
  }

  #pragma unroll
  for (int r = 0; r < 8; ++r) {
    red[w][r * 32 + lane]       = acc0[r];
    red[w][256 + r * 32 + lane] = acc1[r];
  }
  __syncthreads();
  for (int idx = tid; idx < 512; idx += 128) {
    float s = red[0][idx] + red[1][idx] + red[2][idx] + red[3][idx];
    int tile = idx >> 8, r = (idx >> 5) & 7, L = idx & 31;
    int mm = tile * 16 + ((L < 16) ? r : r + 8);
    int nn = nbase + (L & 15);
    // y is re-read by kernels 3/4 (8 MB, fits L2) -> regular temporal store
    y[(size_t)mm * FLAT_OUT + nn] = s + b2[nn];
  }
}

// ---------------------------------------------------------------------------
// Kernel 3 (per batch): G = Y^T Y via WMMA; Cholesky G = R^T R; Rinv = R^{-1}
// grid = 32, block = 256 (8 waves: 4 G-tiles x 2 K-halves)
// ---------------------------------------------------------------------------
__global__ __launch_bounds__(256) void k_cholqr_rinv(
    const float* __restrict__ yws, float* __restrict__ rinv) {
  __shared__ float red[8][256];
  __shared__ float G[32][32];
  __shared__ float RI[32][33];
  const int tid  = threadIdx.x;
  const int w    = tid >> 5, lane = tid & 31;
  const int b    = blockIdx.x;
  const float* Y = yws + (size_t)b * FLAT_OUT;   // [2048][32]
  const int t = w & 3, ti = t >> 1, tj = t & 1;
  const int kstart = (w >> 2) * 1024;
  const int m   = lane & 15;
  const int kkb = (lane >> 4) * 2;

  v8f acc = {};
  for (int k = kstart; k < kstart + 1024; k += 4) {
    v2f a, bb;
    a.x  = Y[(k + kkb) * RANK + ti * 16 + m];      // A = Y^T (16x4)
    a.y  = Y[(k + kkb + 1) * RANK + ti * 16 + m];
    bb.x = Y[(k + kkb) * RANK + tj * 16 + m];      // B = Y (4x16)
    bb.y = Y[(k + kkb + 1) * RANK + tj * 16 + m];
    acc = wmma4(a, bb, acc);
  }
  #pragma unroll
  for (int r = 0; r < 8; ++r) red[w][r * 32 + lane] = acc[r];
  __syncthreads();

  for (int idx = tid; idx < 1024; idx += 256) {
    int tt = idx >> 8, rem = idx & 255;
    float s = red[tt][rem] + red[tt + 4][rem];     // fold the two K-halves
    int r = rem >> 5, L = rem & 31;
    int ii = (tt >> 1) * 16 + ((L < 16) ? r : r + 8);
    int jj = (tt & 1) * 16 + (L & 15);
    G[ii][jj] = s;
  }
  __syncthreads();

  // Right-looking Cholesky on the upper triangle: G -> R with G = R^T R
  for (int k = 0; k < 32; ++k) {
    float gkk = G[k][k];
    __syncthreads();
    float rkk = sqrtf(gkk);
    if (tid < 32) {
      int j = tid;
      if (j > k)       G[k][j] = G[k][j] / rkk;
      else if (j == k) G[k][k] = rkk;
    }
    __syncthreads();
    for (int idx = tid; idx < 1024; idx += 256) {
      int i = idx >> 5, j = idx & 31;
      if (i > k && j >= i) G[i][j] -= G[k][i] * G[k][j];
    }
    __syncthreads();
  }

  // Upper-triangular inverse: solve R * X = I, lane j owns column j
  if (tid < 32) {
    int j = tid;
    RI[j][j] = 1.0f / G[j][j];
    for (int i = j - 1; i >= 0; --i) {
      float s = 0.0f;
      for (int kq = i + 1; kq <= j; ++kq) s += G[i][kq] * RI[kq][j];
      RI[i][j] = -s / G[i][i];
    }
    for (int i = j + 1; i < 32; ++i) RI[i][j] = 0.0f;
  }
  __syncthreads();
  for (int idx = tid; idx < 1024; idx += 256)
    rinv[b * 1024 + idx] = RI[idx >> 5][idx & 31];
}

// ---------------------------------------------------------------------------
// Kernel 4: Q = Y @ Rinv per batch, in place (row block fully loaded before
// stored; stores depend on the final WMMA through the accumulator).
// grid = 32 batches * 32 blocks, block = 128 (4 waves, 1 M-tile each)
// ---------------------------------------------------------------------------
__global__ __launch_bounds__(128) void k_apply_q(
    float* __restrict__ y, const float* __restrict__ rinv) {
  const int tid  = threadIdx.x;
  const int w    = tid >> 5, lane = tid & 31;
  const int b    = blockIdx.x >> 5;
  const int mt   = (blockIdx.x & 31) * 4 + w;    // 0..127 tile of 16 rows
  float* Yb      = y + (size_t)b * FLAT_OUT;
  const float* R = rinv + b * 1024;
  const int m   = lane & 15;
  const int kkb = (lane >> 4) * 2;
  const int nl  = lane & 15;

  v8f acc0 = {}; v8f acc1 = {};
  const float* yrow = Yb + (mt * 16 + m) * RANK;
  #pragma unroll
  for (int k = 0; k < 32; k += 4) {
    v2f a = *(const v2f*)(yrow + k + kkb);
    v2f b0, b1;
    b0.x = R[(k + kkb) * 32 + nl];
    b0.y = R[(k + kkb + 1) * 32 + nl];
    b1.x = R[(k + kkb) * 32 + 16 + nl];
    b1.y = R[(k + kkb + 1) * 32 + 16 + nl];
    acc0 = wmma4(a, b0, acc0);
    acc1 = wmma4(a, b1, acc1);
  }
  #pragma unroll
  for (int r = 0; r < 8; ++r) {
    int mm = mt * 16 + ((lane < 16) ? r : r + 8);
    // Q is never re-read on device -> non-temporal store, don't dirty L2
    __builtin_nontemporal_store(acc0[r], &Yb[mm * RANK + nl]);
    __builtin_nontemporal_store(acc1[r], &Yb[mm * RANK + 16 + nl]);
  }
}

// ---------------------------------------------------------------------------
extern "C" void kernel_launch(void* const* d_in, const int* in_sizes, int n_in,
                              void* d_out, int out_size, void* d_ws,
                              size_t ws_size, hipStream_t stream) {
  (void)in_sizes; (void)n_in; (void)out_size; (void)ws_size;
  const float* x  = (const float*)d_in[0];
  const float* W1 = (const float*)d_in[1];
  const float* b1 = (const float*)d_in[2];
  const float* W2 = (const float*)d_in[3];
  const float* b2 = (const float*)d_in[4];

  char*  ws   = (char*)d_ws;
  float* h    = (float*)ws;                    // 32*2048*4   = 256 KB
  float* rinv = (float*)(ws + (512u << 10));   // 32*32*32*4  = 128 KB
  float* y    = (float*)d_out;                 // intermediate y, rewritten to Q

  k_gemm1_gelu<<<HIDDEN / 16,   128, 0, stream>>>(x, W1, b1, h);
  k_gemm2     <<<FLAT_OUT / 16, 128, 0, stream>>>(h, W2, b2, y);
  k_cholqr_rinv<<<B_SZ,         256, 0, stream>>>(y, rinv);
  k_apply_q   <<<B_SZ * 32,     128, 0, stream>>>(y, rinv);
}